// ConvBlock_29394756173832
// MI455X (gfx1250) — compile-verified
//
#include <hip/hip_runtime.h>

#define NUM_NODES 100000
#define NUM_EDGES 20000
#define NNZ       1600000
#define IN_C      128
#define OUT_C     128

typedef float v2f __attribute__((ext_vector_type(2)));
typedef float v8f __attribute__((ext_vector_type(8)));

// ---------------------------------------------------------------------------
// Kernel 1: degree counting (f32 hardware atomics, L2-resident targets)
// ---------------------------------------------------------------------------
__global__ __launch_bounds__(256) void hg_degrees(const int* __restrict__ hidx,
                                                  float* __restrict__ deg_n,
                                                  float* __restrict__ deg_e) {
    int i = blockIdx.x * 256 + threadIdx.x;
    if (i >= NNZ) return;
    int v = hidx[i];
    int e = hidx[NNZ + i];
    unsafeAtomicAdd(&deg_n[v], 1.0f);
    unsafeAtomicAdd(&deg_e[e], 1.0f);
}

// ---------------------------------------------------------------------------
// Kernel 2: xw = x @ W using fp32 WMMA (V_WMMA_F32_16X16X4_F32).
// One 16x16 output tile per wave; 8 waves per block cover one 16-row strip
// across all 128 output columns. K=128 -> 32 WMMA steps of K=4.
//
// A 16x4 f32 layout (2 VGPRs/lane): lanes 0-15 hold row M=lane, K={0,1};
// lanes 16-31 hold row M=lane-16, K={2,3}. B 4x16 mirrored (row striped
// across lanes). C/D: VGPR r -> M=r (lanes 0-15) / M=8+r (lanes 16-31).
// ---------------------------------------------------------------------------
__global__ __launch_bounds__(256) void hg_gemm_xw(const float* __restrict__ x,
                                                  const float* __restrict__ w,
                                                  float* __restrict__ xw) {
    const int wave = threadIdx.x >> 5;       // 0..7 -> N tile
    const int lane = threadIdx.x & 31;
    const int row0 = blockIdx.x * 16;
    const int col0 = wave * 16;
    const int mn   = lane & 15;              // M for A-rows, N for B-cols
    const int koff = (lane >> 4) << 1;       // 0 (lanes 0-15) or 2 (lanes 16-31)

    const float* __restrict__ xrow = x + (size_t)(row0 + mn) * IN_C;
    v8f acc = {};

#pragma unroll
    for (int k0 = 0; k0 < IN_C; k0 += 4) {
        // A fragment: two consecutive K values for this lane (8B load)
        const float2 af = *(const float2*)(xrow + k0 + koff);
        v2f a; a.x = af.x; a.y = af.y;
        // B fragment: rows (k0+koff, k0+koff+1), column col0+mn
        v2f b;
        b.x = w[(size_t)(k0 + koff    ) * OUT_C + col0 + mn];
        b.y = w[(size_t)(k0 + koff + 1) * OUT_C + col0 + mn];
        acc = __builtin_amdgcn_wmma_f32_16x16x4_f32(false, a, false, b,
                                                    (short)0, acc, false, false);
    }

    const int rbase = (lane >> 4) << 3;      // 0 or 8
#pragma unroll
    for (int r = 0; r < 8; ++r) {
        xw[(size_t)(row0 + rbase + r) * OUT_C + col0 + mn] = acc[r];
    }
}

// ---------------------------------------------------------------------------
// Kernel 3: node -> hyperedge scatter-add (raw sums).
// 32 lanes per incidence; each lane owns 4 channels (float4 gather + 4
// global_atomic_add_f32 into the 10 MB L2-resident edge table).
// ---------------------------------------------------------------------------
__global__ __launch_bounds__(256) void hg_scatter_n2e(const int* __restrict__ hidx,
                                                      const float* __restrict__ xw,
                                                      float* __restrict__ edge_raw) {
    const int t = blockIdx.x * 256 + threadIdx.x;
    const int i = t >> 5;                    // incidence index
    if (i >= NNZ) return;
    const int c = (t & 31) << 2;             // channel base
    const int v = hidx[i];
    const int e = hidx[NNZ + i];
    const float4 val = *(const float4*)(xw + (size_t)v * OUT_C + c);
    float* dst = edge_raw + (size_t)e * OUT_C + c;
    unsafeAtomicAdd(dst + 0, val.x);
    unsafeAtomicAdd(dst + 1, val.y);
    unsafeAtomicAdd(dst + 2, val.z);
    unsafeAtomicAdd(dst + 3, val.w);
}

// ---------------------------------------------------------------------------
// Kernel 4: hyperedge -> node scatter-add with B^{-1} folded into the gather.
// ---------------------------------------------------------------------------
__global__ __launch_bounds__(256) void hg_scatter_e2n(const int* __restrict__ hidx,
                                                      const float* __restrict__ edge_raw,
                                                      const float* __restrict__ deg_e,
                                                      float* __restrict__ out) {
    const int t = blockIdx.x * 256 + threadIdx.x;
    const int i = t >> 5;
    if (i >= NNZ) return;
    const int c = (t & 31) << 2;
    const int v = hidx[i];
    const int e = hidx[NNZ + i];
    const float d = deg_e[e];
    const float binv = (d > 0.0f) ? (1.0f / d) : 0.0f;
    const float4 val = *(const float4*)(edge_raw + (size_t)e * OUT_C + c);
    float* dst = out + (size_t)v * OUT_C + c;
    unsafeAtomicAdd(dst + 0, val.x * binv);
    unsafeAtomicAdd(dst + 1, val.y * binv);
    unsafeAtomicAdd(dst + 2, val.z * binv);
    unsafeAtomicAdd(dst + 3, val.w * binv);
}

// ---------------------------------------------------------------------------
// Kernel 5: out = out * D^{-1} + bias  (in place, float4)
// ---------------------------------------------------------------------------
__global__ __launch_bounds__(256) void hg_finalize(float* __restrict__ out,
                                                   const float* __restrict__ deg_n,
                                                   const float* __restrict__ bias) {
    const int t = blockIdx.x * 256 + threadIdx.x;
    const int node = t >> 5;                 // OUT_C/4 == 32 lanes per node
    if (node >= NUM_NODES) return;
    const int c = (t & 31) << 2;
    const float d = deg_n[node];
    const float dinv = (d > 0.0f) ? (1.0f / d) : 0.0f;
    float4 vv = *(float4*)(out + (size_t)node * OUT_C + c);
    const float4 b = *(const float4*)(bias + c);
    vv.x = vv.x * dinv + b.x;
    vv.y = vv.y * dinv + b.y;
    vv.z = vv.z * dinv + b.z;
    vv.w = vv.w * dinv + b.w;
    *(float4*)(out + (size_t)node * OUT_C + c) = vv;
}

// ---------------------------------------------------------------------------
// Launch
// ---------------------------------------------------------------------------
extern "C" void kernel_launch(void* const* d_in, const int* in_sizes, int n_in,
                              void* d_out, int out_size, void* d_ws, size_t ws_size,
                              hipStream_t stream) {
    const float* x      = (const float*)d_in[0];
    const int*   hidx   = (const int*)d_in[1];   // [2, NNZ]: node row, then edge row
    // d_in[2] = temb (unused by reference forward)
    const float* weight = (const float*)d_in[3];
    const float* bias   = (const float*)d_in[4];
    float*       out    = (float*)d_out;

    // Workspace layout (floats): xw | edge_raw | deg_n | deg_e
    float* xw       = (float*)d_ws;
    float* edge_raw = xw + (size_t)NUM_NODES * OUT_C;
    float* deg_n    = edge_raw + (size_t)NUM_EDGES * OUT_C;
    float* deg_e    = deg_n + NUM_NODES;

    // Zero accumulators (graph-capture safe)
    hipMemsetAsync(edge_raw, 0, sizeof(float) * (size_t)NUM_EDGES * OUT_C, stream);
    hipMemsetAsync(deg_n,    0, sizeof(float) * (size_t)(NUM_NODES + NUM_EDGES), stream);
    hipMemsetAsync(out,      0, sizeof(float) * (size_t)NUM_NODES * OUT_C, stream);

    hg_degrees<<<(NNZ + 255) / 256, 256, 0, stream>>>(hidx, deg_n, deg_e);

    hg_gemm_xw<<<NUM_NODES / 16, 256, 0, stream>>>(x, weight, xw);

    const int scat_blocks = (int)(((size_t)NNZ * 32 + 255) / 256);   // 200000
    hg_scatter_n2e<<<scat_blocks, 256, 0, stream>>>(hidx, xw, edge_raw);
    hg_scatter_e2n<<<scat_blocks, 256, 0, stream>>>(hidx, edge_raw, deg_e, out);

    const int fin_blocks = (int)(((size_t)NUM_NODES * 32 + 255) / 256); // 12500
    hg_finalize<<<fin_blocks, 256, 0, stream>>>(out, deg_n, bias);
}